// DiffGeomPropsApprox_8564164788834
// MI455X (gfx1250) — compile-verified
//
#include <hip/hip_runtime.h>
#include <math.h>

#define M_POINTS 4096
#define N_TILES  (M_POINTS / 16)   // 256
#define KNN 16
#define WG_THREADS 256
#define WAVES_PER_WG 8
#define Q_PER_WG 128   // 8 waves * 16 queries

typedef __attribute__((ext_vector_type(2))) float v2f;
typedef __attribute__((ext_vector_type(8))) float v8f;

__global__ __launch_bounds__(WG_THREADS)
void knn_cov_eig_kernel(const float* __restrict__ X,
                        const float* __restrict__ uv,
                        float* __restrict__ out)
{
    // A-operand staged in exact WMMA lane layout:
    // aS[tile*32 + l]      (l<16)  = (-2*u_j, -2*v_j), j = tile*16 + l   (K=0,1 rows)
    // aS[tile*32 + 16 + l] (l<16)  = (|p_j|^2, 1.0),   j = tile*16 + l   (K=2,3 rows)
    __shared__ float2 aS[N_TILES * 32];              // 64 KB
    __shared__ float  mdS[WAVES_PER_WG][16][KNN];    //  8 KB: merge dists
    __shared__ int    miS[WAVES_PER_WG][16][KNN];    //  8 KB: merge idx

    const int tid   = threadIdx.x;
    const int lane  = tid & 31;
    const int wave  = tid >> 5;
    const int b     = blockIdx.y;
    const int chunk = blockIdx.x;

    // ---- Stage batch uv into LDS, pre-transformed into A-operand layout ----
    const float2* uvb = (const float2*)uv + (size_t)b * M_POINTS;
    for (int t = tid; t < M_POINTS; t += WG_THREADS) {
        float2 p  = uvb[t];
        float  r2 = p.x * p.x + p.y * p.y;
        int tile = t >> 4;
        int l    = t & 15;
        float2 lo; lo.x = -2.0f * p.x; lo.y = -2.0f * p.y;
        float2 hi; hi.x = r2;          hi.y = 1.0f;
        aS[tile * 32 + l]      = lo;
        aS[tile * 32 + 16 + l] = hi;
    }
    __syncthreads();

    // ---- Per-lane query assignment (lane L and L+16 share query L) ----
    const int qloc = lane & 15;
    const int qg   = chunk * Q_PER_WG + wave * 16 + qloc;   // query index in batch

    float2 quv = uvb[qg];                       // one-time global load
    float  r2q = quv.x * quv.x + quv.y * quv.y;

    // B operand (4x16, queries as columns): rows K=0,1 -> (u,v) in lanes 0-15;
    // rows K=2,3 -> (1, |q|^2) in lanes 16-31.
    v2f bop;
    if (lane < 16) { bop.x = quv.x; bop.y = quv.y; }
    else           { bop.x = 1.0f;  bop.y = r2q;  }

    // ---- Per-lane top-16 (ascending) over this lane's candidate half ----
    float dist[KNN];
    int   idx [KNN];
#pragma unroll
    for (int s = 0; s < KNN; ++s) { dist[s] = 3.0e38f; idx[s] = 0; }

    const int half8 = (lane >> 4) << 3;   // 0 for lanes 0-15, 8 for lanes 16-31

    // ---- Software-pipelined candidate sweep ----
    v2f aop;
    {
        float2 t0 = aS[lane];               // tile 0 operand
        aop.x = t0.x; aop.y = t0.y;
    }

#pragma unroll 2
    for (int tile = 0; tile < N_TILES; ++tile) {
        // Prefetch next tile's A operand (wrapped index: always in-range,
        // branch-free). ds_load has a full WMMA+select iteration to complete.
        v2f anext;
        {
            int nt = (tile + 1) & (N_TILES - 1);
            float2 t = aS[nt * 32 + lane];
            anext.x = t.x; anext.y = t.y;
        }

        v8f c = {};
        c = __builtin_amdgcn_wmma_f32_16x16x4_f32(
                false, aop, false, bop, (short)0, c, false, false);

        // Cheap whole-tile reject: min of this lane's 8 candidate distances.
        float m01 = fminf(c[0], c[1]);
        float m23 = fminf(c[2], c[3]);
        float m45 = fminf(c[4], c[5]);
        float m67 = fminf(c[6], c[7]);
        float m   = fminf(fminf(m01, m23), fminf(m45, m67));

        if (m < dist[KNN - 1]) {
            int jbase = tile * 16 + half8;
#pragma unroll
            for (int r = 0; r < 8; ++r) {
                float d = c[r];
                int   j = jbase + r;
                if (d < dist[KNN - 1]) {
                    // branchless sorted insert (ascending)
#pragma unroll
                    for (int s = KNN - 1; s > 0; --s) {
                        bool  sh = dist[s - 1] > d;
                        float nd = sh ? dist[s - 1] : d;
                        int   ni = sh ? idx[s - 1]  : j;
                        if (dist[s] > d) { dist[s] = nd; idx[s] = ni; }
                    }
                    if (dist[0] > d) { dist[0] = d; idx[0] = j; }
                }
            }
        }

        aop = anext;
    }

    // ---- Merge the two half-lists for each query via LDS ----
    __syncthreads();
    if (lane >= 16) {
#pragma unroll
        for (int s = 0; s < KNN; ++s) {
            mdS[wave][qloc][s] = dist[s];
            miS[wave][qloc][s] = idx[s];
        }
    }
    __syncthreads();

    if (lane < 16) {
#pragma unroll
        for (int s2 = 0; s2 < KNN; ++s2) {
            float d = mdS[wave][qloc][s2];
            int   j = miS[wave][qloc][s2];
            if (d < dist[KNN - 1]) {
#pragma unroll
                for (int s = KNN - 1; s > 0; --s) {
                    bool  sh = dist[s - 1] > d;
                    float nd = sh ? dist[s - 1] : d;
                    int   ni = sh ? idx[s - 1]  : j;
                    if (dist[s] > d) { dist[s] = nd; idx[s] = ni; }
                }
                if (dist[0] > d) { dist[0] = d; idx[0] = j; }
            }
        }

        // ---- Gather neighbors, mean-center, 3x3 covariance ----
        const float* Xb = X + (size_t)b * M_POINTS * 3;
        float px[KNN], py[KNN], pz[KNN];
        float sx = 0.f, sy = 0.f, sz = 0.f;
#pragma unroll
        for (int k = 0; k < KNN; ++k) {
            const float* p = Xb + idx[k] * 3;
            px[k] = p[0]; py[k] = p[1]; pz[k] = p[2];
            sx += px[k]; sy += py[k]; sz += pz[k];
        }
        const float inv = 1.0f / (float)KNN;
        float mx = sx * inv, my = sy * inv, mz = sz * inv;

        float a11 = 0.f, a22 = 0.f, a33 = 0.f;
        float a12 = 0.f, a13 = 0.f, a23 = 0.f;
#pragma unroll
        for (int k = 0; k < KNN; ++k) {
            float dx = px[k] - mx, dy = py[k] - my, dz = pz[k] - mz;
            a11 += dx * dx; a22 += dy * dy; a33 += dz * dz;
            a12 += dx * dy; a13 += dx * dz; a23 += dy * dz;
        }

        // ---- Closed-form symmetric 3x3 eigenvalues (descending) ----
        float tr3 = (a11 + a22 + a33) * (1.0f / 3.0f);
        float p1  = a12 * a12 + a13 * a13 + a23 * a23;
        float b11 = a11 - tr3, b22 = a22 - tr3, b33 = a33 - tr3;
        float p2  = b11 * b11 + b22 * b22 + b33 * b33 + 2.0f * p1;
        float pm  = sqrtf(p2 * (1.0f / 6.0f));

        float e1, e2, e3;
        if (pm < 1e-20f) {
            e1 = e2 = e3 = tr3;
        } else {
            float ip  = 1.0f / pm;
            float c11 = b11 * ip, c22 = b22 * ip, c33 = b33 * ip;
            float c12 = a12 * ip, c13 = a13 * ip, c23 = a23 * ip;
            float detB = c11 * (c22 * c33 - c23 * c23)
                       - c12 * (c12 * c33 - c23 * c13)
                       + c13 * (c12 * c23 - c22 * c13);
            float r = 0.5f * detB;
            r = fminf(1.0f, fmaxf(-1.0f, r));
            float phi = acosf(r) * (1.0f / 3.0f);
            const float TWO_PI_3 = 2.0943951023931953f;
            e1 = tr3 + 2.0f * pm * cosf(phi);
            e3 = tr3 + 2.0f * pm * cosf(phi + TWO_PI_3);
            e2 = 3.0f * tr3 - e1 - e3;
        }

        float* o = out + ((size_t)b * M_POINTS + qg) * 3;
        o[0] = e1; o[1] = e2; o[2] = e3;
    }
}

extern "C" void kernel_launch(void* const* d_in, const int* in_sizes, int n_in,
                              void* d_out, int out_size, void* d_ws, size_t ws_size,
                              hipStream_t stream) {
    const float* X  = (const float*)d_in[0];   // (B, 4096, 3) f32
    const float* uv = (const float*)d_in[1];   // (B, 4096, 2) f32
    float* out = (float*)d_out;                // (B, 4096, 3) f32

    int B = in_sizes[0] / (M_POINTS * 3);
    dim3 grid(M_POINTS / Q_PER_WG, B, 1);      // (32, 8)
    dim3 block(WG_THREADS, 1, 1);
    knn_cov_eig_kernel<<<grid, block, 0, stream>>>(X, uv, out);
}